// GINLayer_23673859736036
// MI455X (gfx1250) — compile-verified
//
#include <hip/hip_runtime.h>

#define DFEAT 128
#define BN_EPS 1e-5f

typedef float v2f __attribute__((ext_vector_type(2)));
typedef float v8f __attribute__((ext_vector_type(8)));

// ---------------------------------------------------------------- h = x
__global__ void k_copy(const float* __restrict__ x, float* __restrict__ h, int n4) {
    int i = blockIdx.x * blockDim.x + threadIdx.x;
    if (i < n4) ((float4*)h)[i] = ((const float4*)x)[i];
}

// ------------------------------------------- h[dst] += x[src]  (1 wave/edge)
__global__ void k_scatter(const float* __restrict__ x, const int* __restrict__ src,
                          const int* __restrict__ dst, float* __restrict__ h, int nE) {
    int e = blockIdx.x * 8 + (threadIdx.x >> 5);
    if (e >= nE) return;
    int lane = threadIdx.x & 31;
    int s = src[e], d = dst[e];
    float4 v = ((const float4*)(x + (size_t)s * DFEAT))[lane];   // global_load_b128
    float* hp = h + (size_t)d * DFEAT + lane * 4;
    unsafeAtomicAdd(hp + 0, v.x);                                // global_atomic_add_f32
    unsafeAtomicAdd(hp + 1, v.y);
    unsafeAtomicAdd(hp + 2, v.z);
    unsafeAtomicAdd(hp + 3, v.w);
}

// ------------------------- Out[50000,128] = A[50000,128] @ W[128,128]^T
// One 16x16 output tile per wave; K swept 4 at a time with full-precision
// V_WMMA_F32_16X16X4_F32.  A-frag: lane&15 = M row, (lane>>4)*2 = K sub-offset,
// float2 covers K,K+1.  B = W^T, so B[k][n] = W[n][k]: same addressing pattern
// against row-major W (contiguous float2 per lane).  D: VGPR v -> M = v+8*(lane>>4).
__global__ void k_gemm(const float* __restrict__ A, const float* __restrict__ W,
                       float* __restrict__ Out) {
    const int lane    = threadIdx.x & 31;
    const int wave    = threadIdx.x >> 5;          // 0..7 -> column tile
    const int rowBase = blockIdx.x * 16;
    const int colBase = wave * 16;
    const int l15     = lane & 15;
    const int khalf   = (lane >> 4) * 2;           // 0 or 2
    const float* aPtr = A + (size_t)(rowBase + l15) * DFEAT + khalf;
    const float* bPtr = W + (size_t)(colBase + l15) * DFEAT + khalf;
    v8f acc = {};
#pragma unroll
    for (int k0 = 0; k0 < DFEAT; k0 += 4) {
        v2f a = *(const v2f*)(aPtr + k0);
        v2f b = *(const v2f*)(bPtr + k0);
        acc = __builtin_amdgcn_wmma_f32_16x16x4_f32(
                  false, a, false, b, (short)0, acc, false, false);
    }
    float* oPtr = Out + (size_t)(rowBase + (lane >> 4) * 8) * DFEAT + colBase + l15;
#pragma unroll
    for (int v = 0; v < 8; ++v) oPtr[(size_t)v * DFEAT] = acc[v];
}

// ---------------------------------------------------------------- stats
__global__ void k_zero(float* __restrict__ s) { s[threadIdx.x] = 0.0f; }

// stats[c] = sum over rows, stats[128+c] = sum of squares.  256 thr = 2 rows x 128 cols.
__global__ void k_stats(const float* __restrict__ h, float* __restrict__ stats, int nRows) {
    int c  = threadIdx.x & 127;
    int r0 = threadIdx.x >> 7;
    float s = 0.0f, s2 = 0.0f;
    for (int r = blockIdx.x * 2 + r0; r < nRows; r += gridDim.x * 2) {
        float v = h[(size_t)r * DFEAT + c];
        s  += v;
        s2 += v * v;
    }
    unsafeAtomicAdd(&stats[c], s);
    unsafeAtomicAdd(&stats[DFEAT + c], s2);
}

// --------------------------------- out = relu?((in-mu)*rsqrt(var+eps)*g + be)
__global__ void k_bn(const float* __restrict__ in, const float* __restrict__ stats,
                     const float* __restrict__ g, const float* __restrict__ be,
                     float* __restrict__ out, int nRows, int doRelu) {
    int i = blockIdx.x * blockDim.x + threadIdx.x;          // float4 index
    int total = nRows * (DFEAT / 4);
    if (i >= total) return;
    int c4 = (i & (DFEAT / 4 - 1)) * 4;
    float4 v = ((const float4*)in)[i];
    float inv = 1.0f / (float)nRows;
    float r[4] = {v.x, v.y, v.z, v.w};
#pragma unroll
    for (int j = 0; j < 4; ++j) {
        int c = c4 + j;
        float mu  = stats[c] * inv;
        float var = stats[DFEAT + c] * inv - mu * mu;        // biased variance
        float sc  = rsqrtf(var + BN_EPS) * g[c];
        float o   = (r[j] - mu) * sc + be[c];
        r[j] = doRelu ? fmaxf(o, 0.0f) : o;
    }
    ((float4*)out)[i] = make_float4(r[0], r[1], r[2], r[3]);
}

// ---------------------------------------------------------------- driver
extern "C" void kernel_launch(void* const* d_in, const int* in_sizes, int n_in,
                              void* d_out, int out_size, void* d_ws, size_t ws_size,
                              hipStream_t stream) {
    const float* x   = (const float*)d_in[0];
    const int*   ei  = (const int*)d_in[1];    // [2, E] row-major (int32 per contract)
    const float* W1  = (const float*)d_in[2];
    // b1 = d_in[3], b2 = d_in[7]: per-column constants cancel exactly in
    // training-mode BatchNorm (BN(y+b) == BN(y)), so they are skipped.
    const float* g1  = (const float*)d_in[4];
    const float* be1 = (const float*)d_in[5];
    const float* W2  = (const float*)d_in[6];
    const float* g2  = (const float*)d_in[8];
    const float* be2 = (const float*)d_in[9];
    const float* g3  = (const float*)d_in[10];
    const float* be3 = (const float*)d_in[11];
    float* out = (float*)d_out;

    const int nNodes = in_sizes[0] / DFEAT;    // 50000 (divisible by 16)
    const int nE     = in_sizes[1] / 2;        // 800000
    const int n4     = nNodes * (DFEAT / 4);
    const int* src = ei;
    const int* dst = ei + nE;

    float* h     = (float*)d_ws;               // [nNodes, 128]  (25.6 MB)
    float* stats = h + (size_t)nNodes * DFEAT; // [256]

    // h = x + segment_sum(x[src], dst)
    k_copy   <<<(n4 + 255) / 256, 256, 0, stream>>>(x, h, n4);
    k_scatter<<<(nE + 7) / 8,     256, 0, stream>>>(x, src, dst, h, nE);

    // layer 1: Linear (WMMA f32) -> BN -> ReLU      (out used as ping-pong scratch)
    k_gemm <<<nNodes / 16, 256, 0, stream>>>(h, W1, out);
    k_zero <<<1,   256, 0, stream>>>(stats);
    k_stats<<<500, 256, 0, stream>>>(out, stats, nNodes);
    k_bn   <<<(n4 + 255) / 256, 256, 0, stream>>>(out, stats, g1, be1, h, nNodes, 1);

    // layer 2: Linear (WMMA f32) -> BN -> ReLU
    k_gemm <<<nNodes / 16, 256, 0, stream>>>(h, W2, out);
    k_zero <<<1,   256, 0, stream>>>(stats);
    k_stats<<<500, 256, 0, stream>>>(out, stats, nNodes);
    k_bn   <<<(n4 + 255) / 256, 256, 0, stream>>>(out, stats, g2, be2, h, nNodes, 1);

    // outer BN (no ReLU) -> d_out
    k_zero <<<1,   256, 0, stream>>>(stats);
    k_stats<<<500, 256, 0, stream>>>(h, stats, nNodes);
    k_bn   <<<(n4 + 255) / 256, 256, 0, stream>>>(h, stats, g3, be3, out, nNodes, 0);
}